// InteractionNetwork_81939386073585
// MI455X (gfx1250) — compile-verified
//
#include <hip/hip_runtime.h>

#define D_S 16
#define D_R 8
#define D_E 64
#define D_X 8
#define D_P 64
#define NCLS 10
#define NB 4
#define N_O 2048
#define N_R 16384

typedef float v2f __attribute__((ext_vector_type(2)));
typedef float v8f __attribute__((ext_vector_type(8)));

__device__ __forceinline__ v8f wmma4(v2f a, v2f b, v8f c) {
  // D = A(16x4,f32) * B(4x16,f32) + C(16x16,f32)
  return __builtin_amdgcn_wmma_f32_16x16x4_f32(
      false, a, false, b, (short)0, c, false, false);
}

// Async global->LDS copy, 16B per lane, tracked by ASYNCcnt (no VGPR staging).
__device__ __forceinline__ void async_b128(void* lds, const void* g) {
  asm volatile("global_load_async_to_lds_b128 %0, %1, off"
               :: "v"((unsigned)(uintptr_t)lds), "v"(g)
               : "memory");
}
__device__ __forceinline__ void wait_async() {
#if __has_builtin(__builtin_amdgcn_s_wait_asynccnt)
  __builtin_amdgcn_s_wait_asynccnt(0);
#else
  asm volatile("s_wait_asynccnt 0" ::: "memory");
#endif
}

// ---------------------------------------------------------------------------
// Kernel A: E = W_r @ [O@R_s ; O@R_r ; R_a] + b_r     -> ws (B, D_E, N_R)
// grid = (N_R/64, NB), block = 128 (4 waves); each wave owns 16 relation cols
// Double-buffered LDS fed by async b128 global->LDS DMA.
// ---------------------------------------------------------------------------
#define KA_RT 64
#define KA_KC 64
#define PAD 4

__launch_bounds__(128)
__global__ void kernA(const float* __restrict__ O, const float* __restrict__ Rs,
                      const float* __restrict__ Rr, const float* __restrict__ Ra,
                      const float* __restrict__ Wr, const float* __restrict__ br,
                      float* __restrict__ Eout) {
  __shared__ __align__(16) float sO[2][D_S][KA_KC + PAD];
  __shared__ __align__(16) float sRs[2][KA_KC][KA_RT + PAD];
  __shared__ __align__(16) float sRr[2][KA_KC][KA_RT + PAD];
  __shared__ float sWr[D_E][41];
  __shared__ float sbr[D_E];
  __shared__ float sBm[4][40][17];

  const int tid = threadIdx.x;
  const int lane = tid & 31, wv = tid >> 5;
  const int l16 = lane & 15, hs = lane >> 4;   // half-select (K offset +2)
  const int b = blockIdx.y;
  const int r0 = blockIdx.x * KA_RT;
  const int rg = tid >> 4;            // row group 0..7
  const int c4 = (tid & 15) * 4;      // column (float4 granularity)

  for (int idx = tid; idx < D_E * 40; idx += 128)
    sWr[idx / 40][idx % 40] = Wr[idx];
  for (int idx = tid; idx < D_E; idx += 128) sbr[idx] = br[idx];

  const float* Ob  = O  + (size_t)b * D_S * N_O;
  const float* Rsb = Rs + (size_t)b * N_O * N_R;
  const float* Rrb = Rr + (size_t)b * N_O * N_R;

#define KA_ISSUE(k0, bf)                                                      \
  {                                                                           \
    _Pragma("unroll") for (int j = 0; j < 2; j++)                             \
      async_b128(&sO[bf][j * 8 + rg][c4],                                     \
                 &Ob[(size_t)(j * 8 + rg) * N_O + (k0) + c4]);                \
    _Pragma("unroll") for (int j = 0; j < 8; j++) {                           \
      async_b128(&sRs[bf][j * 8 + rg][c4],                                    \
                 &Rsb[(size_t)((k0) + j * 8 + rg) * N_R + r0 + c4]);          \
      async_b128(&sRr[bf][j * 8 + rg][c4],                                    \
                 &Rrb[(size_t)((k0) + j * 8 + rg) * N_R + r0 + c4]);          \
    }                                                                         \
  }

  v8f c1 = {};  // M1 tile (16 x 16)
  v8f c2 = {};  // M2 tile
  const int nb = wv * 16;

  KA_ISSUE(0, 0);
  wait_async();
  __syncthreads();
  int buf = 0;
  for (int k0 = 0; k0 < N_O; k0 += KA_KC) {
    const int nxt = k0 + KA_KC;
    if (nxt < N_O) KA_ISSUE(nxt, buf ^ 1);    // async DMA next chunk
#pragma unroll
    for (int kk = 0; kk < KA_KC; kk += 4) {   // compute hides DMA latency
      const int kb = kk + 2 * hs;
      v2f a, bs_, br_;
      a.x   = sO[buf][l16][kb];        a.y   = sO[buf][l16][kb + 1];
      bs_.x = sRs[buf][kb][nb + l16];  bs_.y = sRs[buf][kb + 1][nb + l16];
      br_.x = sRr[buf][kb][nb + l16];  br_.y = sRr[buf][kb + 1][nb + l16];
      c1 = wmma4(a, bs_, c1);
      c2 = wmma4(a, br_, c2);
    }
    wait_async();
    __syncthreads();
    buf ^= 1;
  }

  // Per-wave Bmat (40 x 16): rows 0-15 = M1, 16-31 = M2, 32-39 = R_a
#pragma unroll
  for (int i = 0; i < 8; i++) {
    const int m = i + 8 * hs;
    sBm[wv][m][l16]      = c1[i];
    sBm[wv][16 + m][l16] = c2[i];
  }
  if (lane < 16) {
#pragma unroll
    for (int d = 0; d < D_R; d++)
      sBm[wv][32 + d][lane] = Ra[((size_t)b * D_R + d) * N_R + r0 + nb + lane];
  }
  __syncthreads();

  // E tile (64 x 16) = W_r(64x40) @ Bmat(40x16) + b_r
#pragma unroll
  for (int mt = 0; mt < 4; mt++) {
    v8f ce = {};
#pragma unroll
    for (int kk = 0; kk < 40; kk += 4) {
      const int kb = kk + 2 * hs;
      v2f a, bb;
      a.x  = sWr[mt * 16 + l16][kb]; a.y  = sWr[mt * 16 + l16][kb + 1];
      bb.x = sBm[wv][kb][l16];       bb.y = sBm[wv][kb + 1][l16];
      ce = wmma4(a, bb, ce);
    }
#pragma unroll
    for (int i = 0; i < 8; i++) {
      const int e = mt * 16 + i + 8 * hs;
      Eout[((size_t)b * D_E + e) * N_R + r0 + nb + l16] = ce[i] + sbr[e];
    }
  }
}

// ---------------------------------------------------------------------------
// Kernel B: E_hat partials: Ehp[seg,b,e,o] = sum_{r in seg} E[b,e,r]*R_r[b,o,r]
// grid = (N_O/32, NB, 4 k-segments), block = 64 (2 waves)
// Double-buffered LDS fed by async b128 global->LDS DMA.
// ---------------------------------------------------------------------------
#define KB_OT 32
#define KB_KC 64
#define KB_SEG (N_R / 4)

__launch_bounds__(64)
__global__ void kernB(const float* __restrict__ E, const float* __restrict__ Rr,
                      float* __restrict__ Ehp) {
  __shared__ __align__(16) float sE[2][D_E][KB_KC + PAD];
  __shared__ __align__(16) float sR[2][KB_OT][KB_KC + PAD];
  const int tid = threadIdx.x;
  const int lane = tid & 31, wv = tid >> 5;
  const int l16 = lane & 15, hs = lane >> 4;
  const int o0 = blockIdx.x * KB_OT;
  const int b = blockIdx.y;
  const int seg = blockIdx.z;
  const int rbase = seg * KB_SEG;
  const int rg = tid >> 4;            // 0..3
  const int c4 = (tid & 15) * 4;

  const float* Eb  = E  + (size_t)b * D_E * N_R;
  const float* Rrb = Rr + (size_t)b * N_O * N_R;

#define KB_ISSUE(k0, bf)                                                      \
  {                                                                           \
    _Pragma("unroll") for (int j = 0; j < 16; j++)                            \
      async_b128(&sE[bf][j * 4 + rg][c4],                                     \
                 &Eb[(size_t)(j * 4 + rg) * N_R + rbase + (k0) + c4]);        \
    _Pragma("unroll") for (int j = 0; j < 8; j++)                             \
      async_b128(&sR[bf][j * 4 + rg][c4],                                     \
                 &Rrb[(size_t)(o0 + j * 4 + rg) * N_R + rbase + (k0) + c4]);  \
  }

  v8f acc[4] = {};
  const int nb = wv * 16;

  KB_ISSUE(0, 0);
  wait_async();
  __syncthreads();
  int buf = 0;
  for (int k0 = 0; k0 < KB_SEG; k0 += KB_KC) {
    const int nxt = k0 + KB_KC;
    if (nxt < KB_SEG) KB_ISSUE(nxt, buf ^ 1);
#pragma unroll
    for (int kk = 0; kk < KB_KC; kk += 4) {
      const int kb = kk + 2 * hs;
      v2f bb;  // B[k][n] = R_r[o=n][r=k] (transposed read from LDS)
      bb.x = sR[buf][nb + l16][kb]; bb.y = sR[buf][nb + l16][kb + 1];
#pragma unroll
      for (int mt = 0; mt < 4; mt++) {
        v2f a;
        a.x = sE[buf][mt * 16 + l16][kb]; a.y = sE[buf][mt * 16 + l16][kb + 1];
        acc[mt] = wmma4(a, bb, acc[mt]);
      }
    }
    wait_async();
    __syncthreads();
    buf ^= 1;
  }
#pragma unroll
  for (int mt = 0; mt < 4; mt++)
#pragma unroll
    for (int i = 0; i < 8; i++) {
      const int e = mt * 16 + i + 8 * hs;
      Ehp[(((size_t)seg * NB + b) * D_E + e) * N_O + o0 + nb + l16] = acc[mt][i];
    }
}

// ---------------------------------------------------------------------------
// Kernel C: epilogue — reduce partials, C=[O;X;E_hat], phi_o, scores, softmax
// ---------------------------------------------------------------------------
__launch_bounds__(256)
__global__ void kernC(const float* __restrict__ O, const float* __restrict__ X,
                      const float* __restrict__ Ehp,
                      const float* __restrict__ Wo, const float* __restrict__ bo,
                      const float* __restrict__ Ws, const float* __restrict__ bs,
                      float* __restrict__ out) {
  const int gid = blockIdx.x * blockDim.x + threadIdx.x;
  if (gid >= NB * N_O) return;
  const int b = gid / N_O, o = gid % N_O;
  float C[D_S + D_X + D_E];
#pragma unroll
  for (int d = 0; d < D_S; d++) C[d] = O[((size_t)b * D_S + d) * N_O + o];
#pragma unroll
  for (int d = 0; d < D_X; d++) C[D_S + d] = X[((size_t)b * D_X + d) * N_O + o];
  for (int e = 0; e < D_E; e++) {
    float s = 0.f;
#pragma unroll
    for (int seg = 0; seg < 4; seg++)
      s += Ehp[(((size_t)seg * NB + b) * D_E + e) * N_O + o];
    C[D_S + D_X + e] = s;
  }
  float sc[NCLS];
#pragma unroll
  for (int c = 0; c < NCLS; c++) sc[c] = bs[c];
  for (int p = 0; p < D_P; p++) {
    float acc = bo[p];
    for (int f = 0; f < D_S + D_X + D_E; f++)
      acc = fmaf(Wo[p * (D_S + D_X + D_E) + f], C[f], acc);
#pragma unroll
    for (int c = 0; c < NCLS; c++) sc[c] = fmaf(Ws[c * D_P + p], acc, sc[c]);
  }
  float mx = sc[0];
#pragma unroll
  for (int c = 1; c < NCLS; c++) mx = fmaxf(mx, sc[c]);
  float sum = 0.f;
#pragma unroll
  for (int c = 0; c < NCLS; c++) { sc[c] = expf(sc[c] - mx); sum += sc[c]; }
  const float inv = 1.f / sum;
#pragma unroll
  for (int c = 0; c < NCLS; c++)
    out[((size_t)b * NCLS + c) * N_O + o] = sc[c] * inv;
}

// ---------------------------------------------------------------------------
extern "C" void kernel_launch(void* const* d_in, const int* in_sizes, int n_in,
                              void* d_out, int out_size, void* d_ws, size_t ws_size,
                              hipStream_t stream) {
  const float* O  = (const float*)d_in[0];
  const float* Rs = (const float*)d_in[1];
  const float* Rr = (const float*)d_in[2];
  const float* Ra = (const float*)d_in[3];
  const float* X  = (const float*)d_in[4];
  const float* Wr = (const float*)d_in[5];
  const float* br = (const float*)d_in[6];
  const float* Wo = (const float*)d_in[7];
  const float* bo = (const float*)d_in[8];
  const float* Ws = (const float*)d_in[9];
  const float* bs = (const float*)d_in[10];
  float* out = (float*)d_out;

  float* Ew  = (float*)d_ws;                    // E:   NB*D_E*N_R   (16 MB)
  float* Ehp = Ew + (size_t)NB * D_E * N_R;     // Ehat partials: 4*NB*D_E*N_O (8 MB)

  kernA<<<dim3(N_R / KA_RT, NB), 128, 0, stream>>>(O, Rs, Rr, Ra, Wr, br, Ew);
  kernB<<<dim3(N_O / KB_OT, NB, 4), 64, 0, stream>>>(Ew, Rr, Ehp);
  kernC<<<dim3((NB * N_O + 255) / 256), 256, 0, stream>>>(O, X, Ehp, Wo, bo, Ws, bs, out);
}